// KGConv_12240656794085
// MI455X (gfx1250) — compile-verified
//
#include <hip/hip_runtime.h>

#define N_NODES 50000
#define N_EDGES 800000
#define FEAT    128
#define EMB     64
#define NRELS   8
#define IN_DIM  320   // 2*FEAT + EMB

typedef __attribute__((ext_vector_type(2))) float v2f;
typedef __attribute__((ext_vector_type(8))) float v8f;

// ---------------------------------------------------------------------------
// Kernel 1: P[N, 256] = X[N,128] @ V[128,256]
//   V[k][j] = W[j][k]            for j in [0,128)    (src projection)
//   V[k][j] = W[j-128][128+k]    for j in [128,256)  (dst projection)
// One wave computes two 16x16 output tiles, each via 32 chained
// V_WMMA_F32_16X16X4_F32 ops (K=128, 4 per step).
// Fragment layouts per CDNA5 ISA 7.12.2:
//   A 16x4 f32 : lanes 0-15 rows M=0..15 {K=0,K=1}; lanes 16-31 {K=2,K=3}
//   B 4x16 f32 : lanes 0-15 cols N=0..15 {K=0,K=1}; lanes 16-31 {K=2,K=3}
//   C/D 16x16  : VGPR v, lane l -> M = v + 8*(l/16), N = l%16
// ---------------------------------------------------------------------------
__global__ __launch_bounds__(256) void proj_gemm_wmma(
    const float* __restrict__ X, const float* __restrict__ W,
    float* __restrict__ P) {
  const int wave = threadIdx.x >> 5;
  const int lane = threadIdx.x & 31;
  const int half = lane >> 4;      // 0: K=0,1  1: K=2,3
  const int lsub = lane & 15;      // row (A) / col (B) within tile
  const int koff = half * 2;
  const int m0   = blockIdx.x * 16;

  // Preload this lane's A fragments for all 32 K-steps (contiguous b64 loads)
  const float* xrow = X + (size_t)(m0 + lsub) * FEAT + koff;
  v2f a[32];
#pragma unroll
  for (int s = 0; s < 32; ++s) a[s] = *(const v2f*)(xrow + 4 * s);

  // Each wave handles 2 of the 16 column tiles (8 waves * 2 * 16 = 256 cols)
#pragma unroll
  for (int t = 0; t < 2; ++t) {
    const int jbase = (wave * 2 + t) * 16;
    const int j     = jbase + lsub;             // output column 0..255
    // B column j, varying K -> contiguous along a row of W
    const float* wrow = (j < FEAT) ? (W + (size_t)j * IN_DIM)
                                   : (W + (size_t)(j - FEAT) * IN_DIM + FEAT);
    wrow += koff;

    v8f acc = {};
#pragma unroll
    for (int s = 0; s < 32; ++s) {
      v2f bf = *(const v2f*)(wrow + 4 * s);
      acc = __builtin_amdgcn_wmma_f32_16x16x4_f32(
          /*neg_a=*/false, a[s], /*neg_b=*/false, bf,
          /*c_mod=*/(short)0, acc, /*reuse_a=*/false, /*reuse_b=*/false);
    }
#pragma unroll
    for (int v = 0; v < 8; ++v) {
      const int row = m0 + v + half * 8;
      P[(size_t)row * 256 + jbase + lsub] = acc[v];
    }
  }
}

// ---------------------------------------------------------------------------
// Kernel 2: Crel[r][o] = b[o] + sum_k rel_emb[r][k] * W[o][256+k]
// (bias folded in: every edge adds b exactly once in the reference)
// ---------------------------------------------------------------------------
__global__ __launch_bounds__(256) void crel_kernel(
    const float* __restrict__ rel_emb, const float* __restrict__ W,
    const float* __restrict__ b, float* __restrict__ Crel) {
  const int idx = blockIdx.x * blockDim.x + threadIdx.x;
  if (idx >= NRELS * FEAT) return;
  const int r = idx / FEAT;
  const int o = idx % FEAT;
  const float* wr = W + (size_t)o * IN_DIM + 2 * FEAT;
  const float* er = rel_emb + r * EMB;
  float acc = b[o];
#pragma unroll 8
  for (int k = 0; k < EMB; ++k) acc = fmaf(er[k], wr[k], acc);
  Crel[idx] = acc;
}

// ---------------------------------------------------------------------------
// Kernel 3: per-edge gather + atomic scatter into per_rel[(r*N+d)*128].
// One wave per edge; each lane owns a contiguous float4 slice of the 128
// features. unsafeAtomicAdd lowers to hardware global_atomic_add_f32
// (no-return -> STOREcnt path, RMW resolved in L2).
// ---------------------------------------------------------------------------
__global__ __launch_bounds__(256) void edge_scatter(
    const int* __restrict__ src, const int* __restrict__ dst,
    const int* __restrict__ rel, const float* __restrict__ P,
    const float* __restrict__ Crel, float* __restrict__ per_rel) {
  const int tid  = blockIdx.x * blockDim.x + threadIdx.x;
  const int edge = tid >> 5;
  const int lane = tid & 31;
  if (edge >= N_EDGES) return;

  const int s = src[edge];
  const int d = dst[edge];
  const int r = rel[edge];

  const float4 va = ((const float4*)(P + (size_t)s * 256))[lane];
  const float4 vb = ((const float4*)(P + (size_t)d * 256 + FEAT))[lane];
  const float4 vc = ((const float4*)(Crel + (size_t)r * FEAT))[lane];

  float* o = per_rel + ((size_t)r * N_NODES + (size_t)d) * FEAT + lane * 4;
  unsafeAtomicAdd(o + 0, va.x + vb.x + vc.x);
  unsafeAtomicAdd(o + 1, va.y + vb.y + vc.y);
  unsafeAtomicAdd(o + 2, va.z + vb.z + vc.z);
  unsafeAtomicAdd(o + 3, va.w + vb.w + vc.w);
}

// ---------------------------------------------------------------------------
// Kernel 4: out[n,f] = sum_r tanh(per_rel[r,n,f])
// ---------------------------------------------------------------------------
__global__ __launch_bounds__(256) void tanh_reduce(
    const float* __restrict__ per_rel, float* __restrict__ out) {
  const size_t idx = (size_t)blockIdx.x * blockDim.x + threadIdx.x;
  if (idx >= (size_t)N_NODES * FEAT) return;
  float acc = 0.0f;
#pragma unroll
  for (int r = 0; r < NRELS; ++r)
    acc += tanhf(per_rel[(size_t)r * N_NODES * FEAT + idx]);
  out[idx] = acc;
}

// ---------------------------------------------------------------------------
// Workspace layout (bytes):
//   P       : [0, 51'200'000)                N*256*4
//   Crel    : [51'200'000, 51'204'096)       8*128*4
//   per_rel : [51'204'096, 256'004'096)      8*N*128*4
// ---------------------------------------------------------------------------
extern "C" void kernel_launch(void* const* d_in, const int* in_sizes, int n_in,
                              void* d_out, int out_size, void* d_ws,
                              size_t ws_size, hipStream_t stream) {
  const float* X       = (const float*)d_in[0];   // inputs  [N,128]
  const float* rel_emb = (const float*)d_in[1];   // [8,64]
  const float* W       = (const float*)d_in[2];   // [128,320]
  const float* b       = (const float*)d_in[3];   // [128]
  const int*   src     = (const int*)d_in[4];     // [E]
  const int*   dst     = (const int*)d_in[5];     // [E]
  const int*   rel     = (const int*)d_in[6];     // [E]
  float*       out     = (float*)d_out;           // [N,128]

  char*  ws      = (char*)d_ws;
  float* P       = (float*)(ws);
  float* Crel    = (float*)(ws + 51200000);
  float* per_rel = (float*)(ws + 51204096);

  // per_rel accumulates via atomics -> must be zeroed every call
  hipMemsetAsync(per_rel, 0, (size_t)NRELS * N_NODES * FEAT * sizeof(float),
                 stream);

  proj_gemm_wmma<<<N_NODES / 16, 256, 0, stream>>>(X, W, P);
  crel_kernel<<<(NRELS * FEAT + 255) / 256, 256, 0, stream>>>(rel_emb, W, b,
                                                              Crel);
  edge_scatter<<<(N_EDGES * 32) / 256, 256, 0, stream>>>(src, dst, rel, P,
                                                         Crel, per_rel);
  tanh_reduce<<<((size_t)N_NODES * FEAT + 255) / 256, 256, 0, stream>>>(
      per_rel, out);
}